// RNN_52613349376119
// MI455X (gfx1250) — compile-verified
//
#include <hip/hip_runtime.h>
#include <hip/hip_bf16.h>

#define T_LEN 4096
#define D_INP 4096
#define HID   2
#define V_OUT 32000

typedef __attribute__((ext_vector_type(2))) float v2f;
typedef __attribute__((ext_vector_type(8))) float v8f;

#define XS_STRIDE 68   // 272 B rows: 16B-aligned, banks (4*r+c)%64 distinct
#define WS_STRIDE 66   // banks (2*n+k)%64 distinct across N-lanes

// ---------------------------------------------------------------------------
// Stage 1: pre[T,2] = x @ W_ih^T + (b_ih + b_hh), via V_WMMA_F32_16X16X4_F32.
// Block = 256 threads (8 waves); each wave owns a 16-row (timestep) tile,
// so one block covers 128 rows; grid = 4096/128 = 32 blocks.
// x tile [128 x 64] staged in LDS; W_ih k-slice staged as a 16x64 B tile
// whose rows 2..15 are statically zero -> unconditional ds_load_b64 B
// fragments, no EXEC predication in the WMMA loop.
// ---------------------------------------------------------------------------
__global__ __launch_bounds__(256)
void rnn_pre_wmma(const float* __restrict__ x,
                  const float* __restrict__ W_ih,
                  const float* __restrict__ b_ih,
                  const float* __restrict__ b_hh,
                  float* __restrict__ pre)
{
    __shared__ float xs[128 * XS_STRIDE];   // 34,816 B
    __shared__ float wsl[16 * WS_STRIDE];   //  4,224 B (rows >= HID stay zero)

    const int tid  = threadIdx.x;
    const int lane = tid & 31;
    const int wv   = tid >> 5;
    const int t0   = blockIdx.x * 128;

    // zero-fill the B tile once (rows 2..15 provide the N-padding)
    for (int i = tid; i < 16 * WS_STRIDE; i += 256) wsl[i] = 0.0f;

    v8f c = {};                               // 16x16 f32 accumulator

    const int rloc = (lane & 15) + wv * 16;   // A-frag row (local)
    const int koff = (lane >> 4) * 2;         // lanes 16-31 hold K=2,3
    const int ncol = lane & 15;               // B/C column (N)
    const float* __restrict__ aprow = &xs[rloc * XS_STRIDE + koff];
    const float* __restrict__ bprow = &wsl[ncol * WS_STRIDE + koff];

    for (int k0 = 0; k0 < D_INP; k0 += 64) {
        __syncthreads();
        // x[t0..t0+127][k0..k0+63] -> LDS, b128 both sides, coalesced
        for (int i = tid; i < 128 * 16; i += 256) {
            const int r  = i >> 4;
            const int c4 = i & 15;
            const float4 v = *(const float4*)(x + (size_t)(t0 + r) * D_INP + k0 + c4 * 4);
            *(float4*)&xs[r * XS_STRIDE + c4 * 4] = v;
        }
        // W_ih[0:2][k0..k0+63] -> B tile rows 0,1
        if (tid < HID * 64) {
            const int rr = tid >> 6, cc = tid & 63;
            wsl[rr * WS_STRIDE + cc] = W_ih[(size_t)rr * D_INP + k0 + cc];
        }
        if (k0 + 64 < D_INP) { // speculative prefetch of next chunk
            __builtin_prefetch(x + (size_t)(t0 + (tid >> 1)) * D_INP + k0 + 64, 0, 1);
        }
        __syncthreads();

        #pragma unroll
        for (int kb = 0; kb < 16; ++kb) {
            v2f a; a.x = aprow[kb * 4]; a.y = aprow[kb * 4 + 1];
            v2f b; b.x = bprow[kb * 4]; b.y = bprow[kb * 4 + 1];
            // (neg_a, A, neg_b, B, c_mod, C, reuse_a, reuse_b)
            c = __builtin_amdgcn_wmma_f32_16x16x4_f32(false, a, false, b,
                                                      (short)0, c, false, false);
        }
    }

    // C/D layout: VGPR g -> M=g (lanes 0-15) or M=g+8 (lanes 16-31), N=lane%16
    if (ncol < HID) {
        const float bias  = b_ih[ncol] + b_hh[ncol];
        const int   mbase = t0 + wv * 16 + ((lane >> 4) << 3);
        #pragma unroll
        for (int g = 0; g < 8; ++g)
            pre[(size_t)(mbase + g) * HID + ncol] = c[g] + bias;
    }
}

// ---------------------------------------------------------------------------
// Stage 2: serial Elman scan (H=2). Latency-bound; fast tanh via v_exp/v_rcp.
// ---------------------------------------------------------------------------
__device__ __forceinline__ float fast_tanh(float xv)
{
    xv = fminf(9.0f, fmaxf(-9.0f, xv));
    const float e = __builtin_amdgcn_exp2f(xv * 2.8853900817779268f); // e^(2x)
    return (e - 1.0f) * __builtin_amdgcn_rcpf(e + 1.0f);
}

__global__ void rnn_scan(const float* __restrict__ pre,
                         const float* __restrict__ W_hh,
                         float* __restrict__ hs)
{
    if (threadIdx.x != 0) return;
    const float w00 = W_hh[0], w01 = W_hh[1], w10 = W_hh[2], w11 = W_hh[3];
    float h0 = 0.0f, h1 = 0.0f;
    for (int t = 0; t < T_LEN; ++t) {
        const float a0 = pre[2 * t]     + w00 * h0 + w01 * h1;
        const float a1 = pre[2 * t + 1] + w10 * h0 + w11 * h1;
        h0 = fast_tanh(a0);
        h1 = fast_tanh(a1);
        hs[2 * t] = h0; hs[2 * t + 1] = h1;
    }
}

// ---------------------------------------------------------------------------
// Stage 3: logits + fused cross-entropy stats. One block per timestep.
// 512 MB of logit stores dominate (~22 us @ 23.3 TB/s); W_fc (256 KB) is
// L2-resident so the second pass recomputes logits instead of re-reading HBM.
// ---------------------------------------------------------------------------
__global__ __launch_bounds__(256)
void logits_ce(const float* __restrict__ hs,
               const float* __restrict__ W_fc,
               const float* __restrict__ b_fc,
               const int* __restrict__ targets,
               float* __restrict__ out,
               float* __restrict__ nll)
{
    __shared__ float red[256];
    const int t   = blockIdx.x;
    const int tid = threadIdx.x;
    const float h0 = hs[2 * t], h1 = hs[2 * t + 1];
    float* orow = out + (size_t)t * V_OUT;

    float lmax = -3.4e38f;
    for (int v = tid; v < V_OUT; v += 256) {
        const float2 w = *(const float2*)(W_fc + 2 * v);
        const float logit = fmaf(h0, w.x, fmaf(h1, w.y, b_fc[v]));
        orow[v] = logit;
        lmax = fmaxf(lmax, logit);
    }
    red[tid] = lmax; __syncthreads();
    for (int s = 128; s > 0; s >>= 1) {
        if (tid < s) red[tid] = fmaxf(red[tid], red[tid + s]);
        __syncthreads();
    }
    const float rmax = red[0];
    __syncthreads();

    float lsum = 0.0f;
    for (int v = tid; v < V_OUT; v += 256) {
        const float2 w = *(const float2*)(W_fc + 2 * v);
        const float logit = fmaf(h0, w.x, fmaf(h1, w.y, b_fc[v]));
        lsum += __builtin_amdgcn_exp2f((logit - rmax) * 1.4426950408889634f);
    }
    red[tid] = lsum; __syncthreads();
    for (int s = 128; s > 0; s >>= 1) {
        if (tid < s) red[tid] += red[tid + s];
        __syncthreads();
    }
    if (tid == 0) {
        const int tgt = targets[t];
        const float2 w = *(const float2*)(W_fc + 2 * tgt);
        const float lt  = fmaf(h0, w.x, fmaf(h1, w.y, b_fc[tgt]));
        const float lse = rmax + __builtin_amdgcn_logf(red[0]) * 0.6931471805599453f;
        nll[t] = lse - lt;
    }
}

// ---------------------------------------------------------------------------
// Stage 4: deterministic mean of per-row NLL (no float atomics -> bit-stable).
// ---------------------------------------------------------------------------
__global__ __launch_bounds__(256)
void reduce_loss(const float* __restrict__ nll, float* __restrict__ loss)
{
    __shared__ float red[256];
    float s = 0.0f;
    for (int t = threadIdx.x; t < T_LEN; t += 256) s += nll[t];
    red[threadIdx.x] = s; __syncthreads();
    for (int k = 128; k > 0; k >>= 1) {
        if (threadIdx.x < k) red[threadIdx.x] += red[threadIdx.x + k];
        __syncthreads();
    }
    if (threadIdx.x == 0) *loss = red[0] * (1.0f / T_LEN);
}

// ---------------------------------------------------------------------------
extern "C" void kernel_launch(void* const* d_in, const int* in_sizes, int n_in,
                              void* d_out, int out_size, void* d_ws, size_t ws_size,
                              hipStream_t stream)
{
    const float* x    = (const float*)d_in[0];
    const int*   tgt  = (const int*)  d_in[1];
    const float* W_ih = (const float*)d_in[2];
    const float* b_ih = (const float*)d_in[3];
    const float* W_hh = (const float*)d_in[4];
    const float* b_hh = (const float*)d_in[5];
    const float* W_fc = (const float*)d_in[6];
    const float* b_fc = (const float*)d_in[7];

    float* out  = (float*)d_out;                  // [T*V] logits
    float* loss = out + (size_t)T_LEN * V_OUT;    // trailing scalar

    float* pre = (float*)d_ws;                    // T*H
    float* hs  = pre + T_LEN * HID;               // T*H
    float* nll = hs  + T_LEN * HID;               // T

    rnn_pre_wmma<<<T_LEN / 128, 256, 0, stream>>>(x, W_ih, b_ih, b_hh, pre);
    rnn_scan    <<<1,            32, 0, stream>>>(pre, W_hh, hs);
    logits_ce   <<<T_LEN,       256, 0, stream>>>(hs, W_fc, b_fc, tgt, out, nll);
    reduce_loss <<<1,           256, 0, stream>>>(nll, loss);
}